// semiCRF_34153579938262
// MI455X (gfx1250) — compile-verified
//
#include <hip/hip_runtime.h>
#include <cstdint>
#include <cstddef>

typedef __attribute__((ext_vector_type(2))) float v2f;
typedef __attribute__((ext_vector_type(8))) float v8f;

constexpr int NB = 512;    // batch
constexpr int NS = 1024;   // sequence length
constexpr int NT = 8;      // tags (START = NT-1)
constexpr int NL = 8;      // max span length
constexpr float NEGV = -1.0e9f;   // matches reference NEG
constexpr float VLOW = -3.0e38f;  // for max-reductions
constexpr float LN2F = 0.6931471805599453f;

// DPP8 selector for xor-lane permute within groups of 8 (no LDS, no waits)
constexpr unsigned dpp8sel(int m) {
  unsigned s = 0;
  for (int i = 0; i < 8; ++i) s |= (unsigned)((i ^ m) & 7) << (3 * i);
  return s;
}
__device__ __forceinline__ float grp8_max(float v) {
  float t;
  t = __int_as_float(__builtin_amdgcn_mov_dpp8(__float_as_int(v), dpp8sel(1)));
  v = fmaxf(v, t);
  t = __int_as_float(__builtin_amdgcn_mov_dpp8(__float_as_int(v), dpp8sel(2)));
  v = fmaxf(v, t);
  t = __int_as_float(__builtin_amdgcn_mov_dpp8(__float_as_int(v), dpp8sel(4)));
  v = fmaxf(v, t);
  return v;
}
__device__ __forceinline__ float grp8_sum(float v) {
  float t;
  t = __int_as_float(__builtin_amdgcn_mov_dpp8(__float_as_int(v), dpp8sel(1)));
  v += t;
  t = __int_as_float(__builtin_amdgcn_mov_dpp8(__float_as_int(v), dpp8sel(2)));
  v += t;
  t = __int_as_float(__builtin_amdgcn_mov_dpp8(__float_as_int(v), dpp8sel(4)));
  v += t;
  return v;
}
// Raw v_log_f32 (log2) * ln2; inputs are >= 0 sums, log(0) = -inf is the
// correct "zero-probability" limit here, denorm fixups unnecessary.
__device__ __forceinline__ float fast_log(float x) {
  return __builtin_amdgcn_logf(x) * LN2F;
}

// One wave per block; blocks [0, NB/2) = forward scan (2 batches each),
// blocks [NB/2, NB) = backward scan (2 batches each).
// WMMA 16x16x4 F32, K=8 via two accumulating issues.
// Rows M = b*8 + l (b in {0,1} local batch, l = lag), cols N = tag t (0..7 valid).
__global__ __launch_bounds__(32) void semicrf_scan(
    const float* __restrict__ feat, const float* __restrict__ trans,
    float* __restrict__ alphas, float* __restrict__ betas,
    float* __restrict__ zbuf) {
  __shared__ __align__(16) float ring[2][NL][NT];   // alpha/beta history
  __shared__ __align__(16) float fring[2][NL][NT];  // feat row window
  __shared__ __align__(16) float rmax[2][NL];       // per-row maxes (window)

  const int lane = threadIdx.x;
  const bool isFwd = (int)blockIdx.x < (NB / 2);
  const int pair = isFwd ? (int)blockIdx.x : ((int)blockIdx.x - NB / 2);
  const int b0 = pair * 2;

  // A-phase roles: lane -> row M = lane&15 (b = M>>3, l = M&7); k-pair base:
  // lanes 0-15 carry K={0,1} (VGPR0,1), lanes 16-31 carry K={2,3}.
  const int rowM = lane & 15;
  const int ab = rowM >> 3;
  const int al = rowM & 7;
  const int k0 = (lane < 16) ? 0 : 2;
  // C-phase roles: lane -> col N = lane&15 (= tag t, valid < 8), b = lane>>4.
  const int cb = lane >> 4;
  const int ct = lane & 15;
  const int ctc = ct & 7;
  const bool tval = ct < 8;

  const float* fc = feat + (size_t)(b0 + cb) * NS * NT;
  float* alphc = alphas + (size_t)(b0 + cb) * NS * NT;
  float* betac = betas + (size_t)(b0 + cb) * NS * NT;

  // Constant B operand: fwd B[k][n] = exp(trans[n][k]); bwd B[k][n] = exp(trans[k][n]).
  // B 4x16 layout mirrors A: VGPR0 = rows K0 (lanes 0-15) / K2 (lanes 16-31), VGPR1 = K1/K3.
  const float msk = tval ? 1.0f : 0.0f;  // zero padding columns n=8..15
  v2f Bop0, Bop1;
  if (isFwd) {
    Bop0.x = __expf(trans[ctc * NT + k0]) * msk;
    Bop0.y = __expf(trans[ctc * NT + k0 + 1]) * msk;
    Bop1.x = __expf(trans[ctc * NT + k0 + 4]) * msk;
    Bop1.y = __expf(trans[ctc * NT + k0 + 5]) * msk;
  } else {
    Bop0.x = __expf(trans[(k0) * NT + ctc]) * msk;
    Bop0.y = __expf(trans[(k0 + 1) * NT + ctc]) * msk;
    Bop1.x = __expf(trans[(k0 + 4) * NT + ctc]) * msk;
    Bop1.y = __expf(trans[(k0 + 5) * NT + ctc]) * msk;
  }

  // LDS init: NEG rings => invalid spans contribute exp(NEG - M) == 0.
  for (int i = lane; i < 2 * NL * NT; i += 32) (&ring[0][0][0])[i] = NEGV;
  for (int i = lane; i < 2 * NL * NT; i += 32) (&fring[0][0][0])[i] = 0.0f;
  if (lane < 16) (&rmax[0][0])[lane] = VLOW;
  __builtin_amdgcn_wave_barrier();

  if (isFwd) {
    // alpha0 = trans[:, START] + feat[:, 0]
    float f0 = fc[ctc];
    float a0 = trans[ctc * NT + (NT - 1)] + f0;
    if (tval) {
      ring[cb][0][ct] = a0;
      fring[cb][0][ct] = f0;
      alphc[ct] = a0;
    }
    float rv = grp8_max(tval ? a0 : VLOW);
    if (ct == 0) rmax[cb][0] = rv;
    __builtin_amdgcn_wave_barrier();

    float fpend = fc[1 * NT + ctc];  // feat[1], pipelined one step ahead
    for (int p = 1; p < NS; ++p) {
      float featpt = fpend;
      int pn = (p + 1 < NS) ? (p + 1) : p;
      fpend = fc[pn * NT + ctc];  // next row; consumed next iteration
      if (p + 8 < NS) __builtin_prefetch(&fc[(p + 8) * NT], 0, 0);
      // sliding-window max per batch
      const float4* rm = (const float4*)&rmax[0][0];
      float4 q0 = rm[0], q1 = rm[1], q2 = rm[2], q3 = rm[3];
      float M0 = fmaxf(fmaxf(fmaxf(q0.x, q0.y), fmaxf(q0.z, q0.w)),
                       fmaxf(fmaxf(q1.x, q1.y), fmaxf(q1.z, q1.w)));
      float M1 = fmaxf(fmaxf(fmaxf(q2.x, q2.y), fmaxf(q2.z, q2.w)),
                       fmaxf(fmaxf(q3.x, q3.y), fmaxf(q3.z, q3.w)));
      float Ma = ab ? M1 : M0;
      float Mc = cb ? M1 : M0;
      // A operand: exp(alpha[p-1-l, k] - M_b)
      int slot = (p - 1 - al) & 7;
      v2f rlo = *(const v2f*)&ring[ab][slot][k0];
      v2f rhi = *(const v2f*)&ring[ab][slot][k0 + 4];
      v2f A0, A1;
      A0.x = __expf(rlo.x - Ma);
      A0.y = __expf(rlo.y - Ma);
      A1.x = __expf(rhi.x - Ma);
      A1.y = __expf(rhi.y - Ma);
      v8f acc = {0.f, 0.f, 0.f, 0.f, 0.f, 0.f, 0.f, 0.f};
      acc = __builtin_amdgcn_wmma_f32_16x16x4_f32(false, A0, false, Bop0,
                                                  (short)0, acc, false, false);
      acc = __builtin_amdgcn_wmma_f32_16x16x4_f32(false, A1, false, Bop1,
                                                  (short)0, acc, false, false);
      // lse over l (in-lane over the 8 accumulator regs), emission folded in
      float s = acc[0] * __expf(0.5f * featpt);
#pragma unroll
      for (int l2 = 1; l2 < 8; ++l2)
        s += acc[l2] * __expf(0.5f * fring[cb][(p - l2) & 7][ctc]);
      float at = Mc + 0.5f * featpt + fast_log(s);
      float rv2 = grp8_max(tval ? at : VLOW);
      if (tval) {
        fring[cb][p & 7][ct] = featpt;  // only read by this same lane later
        ring[cb][p & 7][ct] = at;
        alphc[p * NT + ct] = at;
      }
      if (ct == 0) rmax[cb][p & 7] = rv2;
      __builtin_amdgcn_wave_barrier();
    }
  } else {
    // beta[S-1] = 0
    float fl = fc[(NS - 1) * NT + ctc];
    if (tval) {
      ring[cb][(NS - 1) & 7][ct] = 0.0f;
      fring[cb][(NS - 1) & 7][ct] = fl;
      betac[(NS - 1) * NT + ct] = 0.0f;
    }
    if (ct == 0) rmax[cb][(NS - 1) & 7] = 0.0f;
    __builtin_amdgcn_wave_barrier();

    float lastb = 0.0f;
    float fpend = fl;  // feat[p+1] for first iteration (p = NS-2)
    for (int p = NS - 2; p >= 0; --p) {
      if (tval) fring[cb][(p + 1) & 7][ct] = fpend;  // feat[p+1]
      int pn = (p > 0) ? p : 0;
      fpend = fc[pn * NT + ctc];  // feat[p]; consumed next iteration
      if (p >= 7) __builtin_prefetch(&fc[(p - 7) * NT], 0, 0);
      __builtin_amdgcn_wave_barrier();
      const float4* rm = (const float4*)&rmax[0][0];
      float4 q0 = rm[0], q1 = rm[1], q2 = rm[2], q3 = rm[3];
      float M0 = fmaxf(fmaxf(fmaxf(q0.x, q0.y), fmaxf(q0.z, q0.w)),
                       fmaxf(fmaxf(q1.x, q1.y), fmaxf(q1.z, q1.w)));
      float M1 = fmaxf(fmaxf(fmaxf(q2.x, q2.y), fmaxf(q2.z, q2.w)),
                       fmaxf(fmaxf(q3.x, q3.y), fmaxf(q3.z, q3.w)));
      float Ma = ab ? M1 : M0;
      float Mc = cb ? M1 : M0;
      // A = exp(beta[p+1+l, k] + 0.5*(feat[p+1,k] + feat[p+1+l,k]) - M_b)
      int slot = (p + 1 + al) & 7;
      v2f rlo = *(const v2f*)&ring[ab][slot][k0];
      v2f rhi = *(const v2f*)&ring[ab][slot][k0 + 4];
      v2f f1lo = *(const v2f*)&fring[ab][(p + 1) & 7][k0];
      v2f f1hi = *(const v2f*)&fring[ab][(p + 1) & 7][k0 + 4];
      v2f f2lo = *(const v2f*)&fring[ab][slot][k0];
      v2f f2hi = *(const v2f*)&fring[ab][slot][k0 + 4];
      v2f A0, A1;
      A0.x = __expf(rlo.x + 0.5f * (f1lo.x + f2lo.x) - Ma);
      A0.y = __expf(rlo.y + 0.5f * (f1lo.y + f2lo.y) - Ma);
      A1.x = __expf(rhi.x + 0.5f * (f1hi.x + f2hi.x) - Ma);
      A1.y = __expf(rhi.y + 0.5f * (f1hi.y + f2hi.y) - Ma);
      v8f acc = {0.f, 0.f, 0.f, 0.f, 0.f, 0.f, 0.f, 0.f};
      acc = __builtin_amdgcn_wmma_f32_16x16x4_f32(false, A0, false, Bop0,
                                                  (short)0, acc, false, false);
      acc = __builtin_amdgcn_wmma_f32_16x16x4_f32(false, A1, false, Bop1,
                                                  (short)0, acc, false, false);
      float s = acc[0] + acc[1] + acc[2] + acc[3] + acc[4] + acc[5] + acc[6] + acc[7];
      float bt = Mc + fast_log(s);
      float rv2 = grp8_max(tval ? bt : VLOW);
      if (tval) {
        ring[cb][p & 7][ct] = bt;
        betac[p * NT + ct] = bt;
      }
      if (ct == 0) rmax[cb][p & 7] = rv2;
      lastb = bt;
      __builtin_amdgcn_wave_barrier();
    }
    // z = lse_t(beta[0] + trans[:, START] + feat[0])
    float val = tval ? (lastb + trans[ctc * NT + (NT - 1)] + fc[ctc]) : VLOW;
    float m8 = grp8_max(val);
    float e = grp8_sum(__expf(val - m8));
    float z = m8 + fast_log(e);
    if (ct == 0) zbuf[b0 + cb] = z;
  }
}

__global__ __launch_bounds__(256) void semicrf_finalize(
    const float* __restrict__ betas, const float* __restrict__ zbuf,
    float* __restrict__ out) {
  int idx = blockIdx.x * 256 + threadIdx.x;
  if (idx >= NB * NS * NT) return;
  int b = idx / (NS * NT);
  out[idx] = __expf(out[idx] + betas[idx] - zbuf[b]);
}

extern "C" void kernel_launch(void* const* d_in, const int* in_sizes, int n_in,
                              void* d_out, int out_size, void* d_ws, size_t ws_size,
                              hipStream_t stream) {
  (void)in_sizes; (void)n_in; (void)out_size; (void)ws_size;
  const float* feat = (const float*)d_in[0];
  const float* trans = (const float*)d_in[1];
  float* out = (float*)d_out;                 // alphas staged here, then overwritten
  float* betas = (float*)d_ws;                // NB*NS*NT floats
  float* zbuf = betas + (size_t)NB * NS * NT; // NB floats

  semicrf_scan<<<dim3(NB), dim3(32), 0, stream>>>(feat, trans, out, betas, zbuf);
  int total = NB * NS * NT;
  semicrf_finalize<<<dim3((total + 255) / 256), dim3(256), 0, stream>>>(betas, zbuf, out);
}